// ModelNew_17411797418163
// MI455X (gfx1250) — compile-verified
//
#include <hip/hip_runtime.h>

// Fused residual-add + RMSNorm + dual SmoothQuant int8 dynamic quant.
// B=4, S=4096, N=4096 fp32. Memory-bound streaming kernel (~1.54 GB traffic,
// ~66us floor @ 23.3 TB/s). Uses gfx1250 async global->LDS loads (ASYNCcnt)
// for software-pipelined staging (marked non-temporal: x1/x2 are read-once),
// and non-temporal stores for the ~1GB of write-once output, preserving the
// 192MB L2 for the gamma/smooth-scale vectors reused by every row.

typedef float v4f __attribute__((ext_vector_type(4)));

#define NCOLS    4096
#define HALF     2048     // elements per pipeline stage (half row)
#define NTHREADS 256
#define RPB      8        // rows per block

__device__ __forceinline__ void async_b128(unsigned ldsOff, const float* g) {
  // GLOBAL_LOAD_ASYNC_TO_LDS_B128: VDST = per-lane LDS byte offset,
  // VADDR = per-lane 64-bit global address. Tracked by ASYNCcnt.
  // th:TH_LOAD_NT -> read-once stream, do not retain in L2.
  asm volatile("global_load_async_to_lds_b128 %0, %1, off th:TH_LOAD_NT"
               :: "v"(ldsOff), "v"((unsigned long long)(const void*)g)
               : "memory");
}

__global__ __launch_bounds__(NTHREADS)
void fused_add_rmsnorm_dualquant(const float* __restrict__ x1,
                                 const float* __restrict__ x2,
                                 const float* __restrict__ gamma,
                                 const float* __restrict__ s1,
                                 const float* __restrict__ s2,
                                 float* __restrict__ out,
                                 int nrows)
{
  __shared__ __align__(16) float sb1[2][HALF];   // x1 staging, double buffered
  __shared__ __align__(16) float sb2[2][HALF];   // x2 staging, double buffered
  __shared__ float red[24];                      // 8 waves x {ssq, m1, m2}

  const int tid = threadIdx.x;
  const size_t BSN = (size_t)nrows * NCOLS;
  float* __restrict__ oXsum  = out;
  float* __restrict__ oYnorm = out + BSN;
  float* __restrict__ oY1    = out + 2 * BSN;
  float* __restrict__ oS1    = out + 3 * BSN;
  float* __restrict__ oY2    = out + 3 * BSN + (size_t)nrows;
  float* __restrict__ oS2    = out + 4 * BSN + (size_t)nrows;

  // Persistent per-thread column data: chunk c covers col = c*1024 + tid*4.
  v4f ga[4], g1v[4], g2v[4];
#pragma unroll
  for (int c = 0; c < 4; ++c) {
    const int col = c * 1024 + tid * 4;
    v4f g = *(const v4f*)(gamma + col);
    ga[c]  = g;
    g1v[c] = g * (*(const v4f*)(s1 + col));
    g2v[c] = g * (*(const v4f*)(s2 + col));
  }

  // Low 32 bits of a generic shared-space address == LDS byte offset.
  const unsigned lb1 = (unsigned)(unsigned long long)(void*)&sb1[0][0] + (unsigned)tid * 16u;
  const unsigned lb2 = (unsigned)(unsigned long long)(void*)&sb2[0][0] + (unsigned)tid * 16u;

  const int row0 = blockIdx.x * RPB;
  const int totalStages = RPB * 2;

  // Issue one pipeline stage: 4 async b128 loads per lane (2 for x1, 2 for x2).
  auto issue = [&](int gs) {
    const int r = row0 + (gs >> 1);
    const size_t base = ((size_t)r << 12) + (size_t)((gs & 1) * HALF) + (size_t)tid * 4;
    const unsigned buf = (unsigned)(gs & 1) * (HALF * 4u);
    async_b128(lb1 + buf,         x1 + base);
    async_b128(lb1 + buf + 4096u, x1 + base + 1024);
    async_b128(lb2 + buf,         x2 + base);
    async_b128(lb2 + buf + 4096u, x2 + base + 1024);
  };

  issue(0);  // prologue

  for (int i = 0; i < RPB; ++i) {
    float ssq = 0.f, m1 = 0.f, m2 = 0.f;
    v4f xs[4];

#pragma unroll
    for (int h = 0; h < 2; ++h) {
      const int gs = i * 2 + h;
      if (gs + 1 < totalStages) {
        issue(gs + 1);                                    // prefetch next stage
        asm volatile("s_wait_asynccnt 0x4" ::: "memory"); // current stage done
      } else {
        asm volatile("s_wait_asynccnt 0x0" ::: "memory");
      }
      const int buf = gs & 1;
      // Each lane reads back exactly the bytes its own async loads wrote:
      // no barrier needed (per-wave ASYNCcnt + same-wave ownership).
#pragma unroll
      for (int c = 0; c < 2; ++c) {
        const int cc = h * 2 + c;
        v4f a = *(const v4f*)&sb1[buf][c * 1024 + tid * 4];
        v4f b = *(const v4f*)&sb2[buf][c * 1024 + tid * 4];
        v4f sx = a + b;
        xs[cc] = sx;
        ssq = fmaf(sx.x, sx.x, ssq);
        ssq = fmaf(sx.y, sx.y, ssq);
        ssq = fmaf(sx.z, sx.z, ssq);
        ssq = fmaf(sx.w, sx.w, ssq);
        v4f t1 = sx * g1v[cc];
        m1 = fmaxf(m1, fmaxf(fmaxf(__builtin_fabsf(t1.x), __builtin_fabsf(t1.y)),
                             fmaxf(__builtin_fabsf(t1.z), __builtin_fabsf(t1.w))));
        v4f t2 = sx * g2v[cc];
        m2 = fmaxf(m2, fmaxf(fmaxf(__builtin_fabsf(t2.x), __builtin_fabsf(t2.y)),
                             fmaxf(__builtin_fabsf(t2.z), __builtin_fabsf(t2.w))));
      }
    }

    // wave32 butterfly reduction, then combine 8 wave partials via LDS.
#pragma unroll
    for (int off = 16; off > 0; off >>= 1) {
      ssq += __shfl_xor(ssq, off, 32);
      m1 = fmaxf(m1, __shfl_xor(m1, off, 32));
      m2 = fmaxf(m2, __shfl_xor(m2, off, 32));
    }
    const int wave = tid >> 5;
    if ((tid & 31) == 0) {
      red[wave] = ssq; red[8 + wave] = m1; red[16 + wave] = m2;
    }
    __syncthreads();
    float tssq = 0.f, tm1 = 0.f, tm2 = 0.f;
#pragma unroll
    for (int w = 0; w < 8; ++w) {
      tssq += red[w];
      tm1 = fmaxf(tm1, red[8 + w]);
      tm2 = fmaxf(tm2, red[16 + w]);
    }
    __syncthreads();  // red[] reused next row

    const float invRms = rsqrtf(tssq * (1.0f / NCOLS) + 1e-5f);
    // q = clip(round(a_raw * 127/max|a_raw|)) : inv_rms cancels in quantizer.
    const float r1 = 127.0f / tm1;
    const float r2 = 127.0f / tm2;

    const int row = row0 + i;
    if (tid == 0) {
      __builtin_nontemporal_store(tm1 * invRms * (1.0f / 127.0f), oS1 + row);
      __builtin_nontemporal_store(tm2 * invRms * (1.0f / 127.0f), oS2 + row);
    }

    const size_t rb = (size_t)row << 12;
#pragma unroll
    for (int c = 0; c < 4; ++c) {
      const size_t idx = rb + (size_t)(c * 1024 + tid * 4);
      v4f sx = xs[c];
      __builtin_nontemporal_store(sx, (v4f*)(oXsum + idx));
      v4f yn = sx * ga[c] * invRms;
      __builtin_nontemporal_store(yn, (v4f*)(oYnorm + idx));
      v4f a1 = sx * g1v[c] * r1;
      v4f q1;
      q1.x = fminf(fmaxf(__builtin_rintf(a1.x), -128.f), 127.f);
      q1.y = fminf(fmaxf(__builtin_rintf(a1.y), -128.f), 127.f);
      q1.z = fminf(fmaxf(__builtin_rintf(a1.z), -128.f), 127.f);
      q1.w = fminf(fmaxf(__builtin_rintf(a1.w), -128.f), 127.f);
      __builtin_nontemporal_store(q1, (v4f*)(oY1 + idx));
      v4f a2 = sx * g2v[c] * r2;
      v4f q2;
      q2.x = fminf(fmaxf(__builtin_rintf(a2.x), -128.f), 127.f);
      q2.y = fminf(fmaxf(__builtin_rintf(a2.y), -128.f), 127.f);
      q2.z = fminf(fmaxf(__builtin_rintf(a2.z), -128.f), 127.f);
      q2.w = fminf(fmaxf(__builtin_rintf(a2.w), -128.f), 127.f);
      __builtin_nontemporal_store(q2, (v4f*)(oY2 + idx));
    }
  }
}

extern "C" void kernel_launch(void* const* d_in, const int* in_sizes, int n_in,
                              void* d_out, int out_size, void* d_ws, size_t ws_size,
                              hipStream_t stream) {
  const float* x1    = (const float*)d_in[0];
  const float* x2    = (const float*)d_in[1];
  const float* gamma = (const float*)d_in[2];
  const float* s1    = (const float*)d_in[3];
  const float* s2    = (const float*)d_in[4];
  float* out = (float*)d_out;

  const int nrows  = in_sizes[0] / NCOLS;  // B*S = 16384
  const int blocks = nrows / RPB;          // 2048
  fused_add_rmsnorm_dualquant<<<dim3(blocks), dim3(NTHREADS), 0, stream>>>(
      x1, x2, gamma, s1, s2, out, nrows);
}